// CppRAConv_44332652430043
// MI455X (gfx1250) — compile-verified
//
#include <hip/hip_runtime.h>

// Problem constants (match reference)
#define CIN   64
#define COUT  64
#define KW    3
#define H     112
#define W     112
#define NIMG  8
#define P     (H * W)          // 12544 patches per image
#define D     (CIN * KW * KW)  // 576
#define MEMSZ 1025
#define REPROWS 1040           // MEMSZ padded up to multiple of 16 -> branch-free stores
#define MIN_SUMMARY (-512)
#define SCALEF 100.0f

typedef float v2f __attribute__((ext_vector_type(2)));
typedef float v8f __attribute__((ext_vector_type(8)));

// ---------------------------------------------------------------------------
// Kernel 1: channel-sum image  S[n,h,w] = sum_c fmap[n,c,h,w]
// ---------------------------------------------------------------------------
__global__ void k_chansum(const float* __restrict__ fmap, float* __restrict__ S) {
  int t = blockIdx.x * blockDim.x + threadIdx.x;
  if (t >= NIMG * P) return;
  int n = t / P, hw = t % P;
  const float* base = fmap + (size_t)n * CIN * P + hw;
  float s = 0.f;
#pragma unroll
  for (int c = 0; c < CIN; ++c) s += base[(size_t)c * P];
  S[t] = s;
}

// ---------------------------------------------------------------------------
// Kernel 2: init first-occurrence table to sentinel
// ---------------------------------------------------------------------------
__global__ void k_init_first(int* __restrict__ first) {
  int t = blockIdx.x * blockDim.x + threadIdx.x;
  if (t < NIMG * MEMSZ) first[t] = 0x7FFFFFFF;
}

// ---------------------------------------------------------------------------
// Kernel 3: per-patch bin (quantized mean) + scatter-min first occurrence
// ---------------------------------------------------------------------------
__global__ void k_bins(const float* __restrict__ S, int* __restrict__ bins,
                       int* __restrict__ first) {
  int t = blockIdx.x * blockDim.x + threadIdx.x;
  if (t >= NIMG * P) return;
  int n = t / P, p = t % P;
  int h = p / W, w = p % W;
  const float* Sn = S + (size_t)n * P;
  float s = 0.f;
#pragma unroll
  for (int kh = -1; kh <= 1; ++kh) {
#pragma unroll
    for (int kw = -1; kw <= 1; ++kw) {
      int hh = h + kh, ww = w + kw;
      if (hh >= 0 && hh < H && ww >= 0 && ww < W) s += Sn[hh * W + ww];
    }
  }
  float mean = s / (float)D;
  int summary = (int)(mean * SCALEF);          // trunc toward zero == astype(int32)
  int b = summary - MIN_SUMMARY;
  b = b < 0 ? 0 : (b > MEMSZ - 1 ? MEMSZ - 1 : b);
  bins[t] = b;
  atomicMin(&first[n * MEMSZ + b], p);
}

// ---------------------------------------------------------------------------
// Kernel 4: WMMA GEMM over representative rows only.
//   rep_out[n, bin, o] = patch(first[n,bin]) . weight[o,:] + bias[o]
// Block = 128 threads (4 waves). blockIdx.x = 16-bin row block, blockIdx.y = n.
// Wave w computes the 16-wide Cout slice [16w, 16w+16).
// ---------------------------------------------------------------------------
#define LDS_STRIDE 580   // 576 + 4 dwords pad: stride % 64 == 4 -> conflict-free A reads

__global__ void __launch_bounds__(128)
k_gemm(const float* __restrict__ fmap, const float* __restrict__ weight,
       const float* __restrict__ bias, const int* __restrict__ first,
       float* __restrict__ rep_out) {
  __shared__ float ap[16 * LDS_STRIDE];
  __shared__ int qrow[16];

  const int n   = blockIdx.y;
  const int rb  = blockIdx.x;        // bins rb*16 .. rb*16+15
  const int tid = threadIdx.x;

  if (tid < 16) {
    int b = rb * 16 + tid;
    int q = (b < MEMSZ) ? first[n * MEMSZ + b] : 0x7FFFFFFF;
    if (q == 0x7FFFFFFF) q = 0;      // unused bin: compute a dummy row (never read)
    qrow[tid] = q;
  }
  __syncthreads();

  // Cooperative gather of the 16 x 576 representative-patch tile into LDS.
  const float* fb = fmap + (size_t)n * CIN * P;
  for (int idx = tid; idx < 16 * D; idx += 128) {
    int r = idx / D, d = idx - r * D;
    int q = qrow[r];
    int h = q / W, w = q - h * W;
    int c = d / 9, rem = d - c * 9;
    int kh = rem / 3, kw = rem - kh * 3;
    int hh = h + kh - 1, ww = w + kw - 1;
    float v = 0.f;
    if (hh >= 0 && hh < H && ww >= 0 && ww < W)
      v = fb[(size_t)c * P + hh * W + ww];
    ap[r * LDS_STRIDE + d] = v;
  }
  __syncthreads();

  const int lane  = tid & 31;
  const int wave  = tid >> 5;
  const int half  = lane >> 4;       // K sub-pair selector (ISA A/B 16x4 layout)
  const int l16   = lane & 15;       // A: row M;  B: col N
  const int ocol  = wave * 16 + l16; // this lane's Cout column

  v8f acc;
  {
    float bv = bias[ocol];           // N is constant per lane across all 8 C VGPRs
#pragma unroll
    for (int i = 0; i < 8; ++i) acc[i] = bv;
  }

  const float* arow = ap + l16 * LDS_STRIDE + half * 2;       // LDS
  const float* brow = weight + (size_t)ocol * D + half * 2;   // global (L2-resident)

#pragma unroll 4
  for (int k0 = 0; k0 < D; k0 += 4) {
    v2f a = *(const v2f*)(arow + k0);   // A[M=l16][k0+2*half .. +1]
    v2f b = *(const v2f*)(brow + k0);   // B[k0+2*half .. +1][N=l16]
    acc = __builtin_amdgcn_wmma_f32_16x16x4_f32(
        /*neg_a=*/false, a, /*neg_b=*/false, b,
        /*c_mod=*/(short)0, acc, /*reuse_a=*/false, /*reuse_b=*/false);
  }

  // C/D layout: VGPR v, lane l -> M = v + 8*(l>>4), N = l&15.
  // rep_out rows padded to REPROWS (multiple of 16) -> no bounds check needed.
  float* ro = rep_out + (size_t)n * REPROWS * COUT;
#pragma unroll
  for (int v = 0; v < 8; ++v) {
    int row = rb * 16 + v + half * 8;
    ro[(size_t)row * COUT + ocol] = acc[v];
  }
}

// ---------------------------------------------------------------------------
// Kernel 5: gather final output  out[n,o,p] = rep_out[n, bins[n,p], o]
// ---------------------------------------------------------------------------
__global__ void k_gather(const int* __restrict__ bins,
                         const float* __restrict__ rep_out,
                         float* __restrict__ out) {
  int t = blockIdx.x * blockDim.x + threadIdx.x;
  if (t >= NIMG * P) return;
  int n = t / P, p = t - n * P;
  const float* row = rep_out + ((size_t)n * REPROWS + bins[t]) * COUT;
  float* ob = out + (size_t)n * COUT * P + p;
#pragma unroll
  for (int o = 0; o < COUT; o += 4) {
    float4 v = *(const float4*)(row + o);
    ob[(size_t)(o + 0) * P] = v.x;
    ob[(size_t)(o + 1) * P] = v.y;
    ob[(size_t)(o + 2) * P] = v.z;
    ob[(size_t)(o + 3) * P] = v.w;
  }
}

// ---------------------------------------------------------------------------
// Launch
// ---------------------------------------------------------------------------
extern "C" void kernel_launch(void* const* d_in, const int* in_sizes, int n_in,
                              void* d_out, int out_size, void* d_ws, size_t ws_size,
                              hipStream_t stream) {
  const float* fmap   = (const float*)d_in[0];  // [8,64,112,112]
  const float* weight = (const float*)d_in[1];  // [64,576]
  const float* bias   = (const float*)d_in[2];  // [64]
  float* out = (float*)d_out;                   // [8,64,112,112]

  // Workspace layout (bytes), all 16B-aligned:
  //   S       : NIMG*P floats           =   401408
  //   bins    : NIMG*P ints             =   401408
  //   first   : NIMG*MEMSZ ints         =    32800
  //   rep_out : NIMG*REPROWS*COUT float =  2129920   (total ~2.97 MB)
  char* ws = (char*)d_ws;
  float* S     = (float*)(ws);
  int*   bins  = (int*)  (ws + 401408);
  int*   first = (int*)  (ws + 802816);
  float* rep   = (float*)(ws + 835616);

  const int tot = NIMG * P;  // 100352
  k_chansum   <<<(tot + 255) / 256, 256, 0, stream>>>(fmap, S);
  k_init_first<<<(NIMG * MEMSZ + 255) / 256, 256, 0, stream>>>(first);
  k_bins      <<<(tot + 255) / 256, 256, 0, stream>>>(S, bins, first);
  dim3 g(REPROWS / 16, NIMG);   // 65 x 8 row-blocks
  k_gemm      <<<g, 128, 0, stream>>>(fmap, weight, bias, first, rep);
  k_gather    <<<(tot + 255) / 256, 256, 0, stream>>>(bins, rep, out);
}